// xNN_31353261261158
// MI455X (gfx1250) — compile-verified
//
#include <hip/hip_runtime.h>

typedef __attribute__((ext_vector_type(2))) float v2f;
typedef __attribute__((ext_vector_type(8))) float v8f;

#define ALPHA 0.2f
#define BATCH 8
#define FDIM  64
#define ROWP  68   // LDS row stride in floats (64 + 4 pad -> conflict-free)

// -------------------------------------------------------------------------
// Kernel 1: dense projection via V_WMMA_F32_16X16X4_F32, LDS-staged.
//   z[n*8+b]   = dot(x[b,n,0:64], W0) + b0
//   acc[n*8+b] = ALPHA * z[n*8+b]
// One wave per (batch b, 16-node tile). The 16x64 f32 tile is contiguous in
// global memory, so each wave streams it with 8 coalesced b128 loads into a
// padded LDS region, then builds A-fragments (ISA 7.12.2 layout: lanes 0-15
// hold K0/K1, lanes 16-31 hold K2/K3) with conflict-free ds_load_b64 reads.
// B = W0 column broadcast across all 16 N-columns, so every column of D is
// the same z vector; lanes 0 and 16 extract it.
// -------------------------------------------------------------------------
__global__ __launch_bounds__(256)
void k_dense_wmma(const float* __restrict__ x, const float* __restrict__ W0,
                  const float* __restrict__ b0, float* __restrict__ z,
                  float* __restrict__ acc, int n_nodes) {
  __shared__ float lds[8][16 * ROWP];          // ~34.8 KB per block

  const int wv     = threadIdx.x >> 5;         // wave in block (0..7)
  const int lane   = threadIdx.x & 31;
  const int tilesN = n_nodes >> 4;             // N is a multiple of 16
  const int tiles  = BATCH * tilesN;

  int t = blockIdx.x * 8 + wv;                 // global tile id
  const bool live = (t < tiles);
  if (!live) t = 0;                            // clamp: keep wave in lockstep
  const int nt = t % tilesN;
  const int b  = t / tilesN;
  const int n0 = nt << 4;

  // ---- stage the contiguous 16x64 tile into padded LDS (coalesced b128) ----
  float* tile = &lds[wv][0];
  const float* gsrc = x + ((size_t)b * n_nodes + (size_t)n0) * FDIM;
#pragma unroll
  for (int i = 0; i < 8; ++i) {
    const int fi = i * 128 + lane * 4;         // flat float index 0..1023
    const float4 v = *(const float4*)(gsrc + fi);
    const int r = fi >> 6;                     // tile row (node)
    const int c = fi & 63;                     // feature
    *(float4*)(tile + r * ROWP + c) = v;
  }
  __syncthreads();

  // ---- WMMA accumulation over F=64 in 16 steps of K=4 ----
  const int row   = lane & 15;                 // M index
  const int khalf = (lane >> 4) << 1;          // 0 (lanes 0-15) / 2 (16-31)
  const float* myrow = tile + row * ROWP + khalf;

  v8f c = {};
#pragma unroll
  for (int s = 0; s < 16; ++s) {
    const int k0 = s * 4;
    v2f a;                                     // A: VGPR0=K(khalf), VGPR1=K(khalf+1)
    a.x = myrow[k0];
    a.y = myrow[k0 + 1];
    v2f w;                                     // B: same K split, uniform over N
    w.x = W0[k0 + khalf];
    w.y = W0[k0 + khalf + 1];
    c = __builtin_amdgcn_wmma_f32_16x16x4_f32(false, a, false, w, (short)0, c,
                                              false, false);
  }

  const float bias = b0[0];
  if (live && row == 0) {                      // lanes 0 (M=0..7) and 16 (M=8..15)
    const int mbase = (lane >> 4) << 3;
#pragma unroll
    for (int r = 0; r < 8; ++r) {
      const float zv   = c[r] + bias;
      const size_t idx = (size_t)(n0 + mbase + r) * BATCH + b;
      z[idx]   = zv;
      acc[idx] = ALPHA * zv;
    }
  }
}

// -------------------------------------------------------------------------
// Kernel 2: SpMM scatter.  acc[dst, :] += 0.8 * w_e * z[src, :]
// One thread per edge; z/acc are [N, 8] so each edge touches one 32B chunk.
// acc (3.2 MB) is L2-resident; hardware FP32 atomics do the reduction.
// -------------------------------------------------------------------------
__global__ __launch_bounds__(256)
void k_spmm(const int* __restrict__ src, const int* __restrict__ dst,
            const float* __restrict__ ew, const float* __restrict__ z,
            float* __restrict__ acc, int n_edges) {
  const int e = blockIdx.x * blockDim.x + threadIdx.x;
  if (e >= n_edges) return;
  const float w = (1.0f - ALPHA) * ew[e];
  const int   s = src[e];
  const int   d = dst[e];
  const float4* zp = (const float4*)(z + (size_t)s * BATCH);
  const float4 z0 = zp[0];
  const float4 z1 = zp[1];
  float* ap = acc + (size_t)d * BATCH;
  unsafeAtomicAdd(ap + 0, w * z0.x);
  unsafeAtomicAdd(ap + 1, w * z0.y);
  unsafeAtomicAdd(ap + 2, w * z0.z);
  unsafeAtomicAdd(ap + 3, w * z0.w);
  unsafeAtomicAdd(ap + 4, w * z1.x);
  unsafeAtomicAdd(ap + 5, w * z1.y);
  unsafeAtomicAdd(ap + 6, w * z1.z);
  unsafeAtomicAdd(ap + 7, w * z1.w);
}

// -------------------------------------------------------------------------
// Kernel 3: 1x1 "dense" for layers 2/3 + re-init accumulator.
//   zi = W*acc[i] + b ; z[i] = zi ; acc[i] = ALPHA*zi
// -------------------------------------------------------------------------
__global__ __launch_bounds__(256)
void k_scale(const float* __restrict__ Wp, const float* __restrict__ bp,
             float* __restrict__ z, float* __restrict__ acc, int count) {
  const int i = blockIdx.x * blockDim.x + threadIdx.x;
  if (i >= count) return;
  const float w  = Wp[0];
  const float bb = bp[0];
  const float zi = w * acc[i] + bb;
  z[i]   = zi;
  acc[i] = ALPHA * zi;
}

// -------------------------------------------------------------------------
// Kernel 4: out[n] = relu(mean_b acc[n,b]) + 0.001
// -------------------------------------------------------------------------
__global__ __launch_bounds__(256)
void k_final(const float* __restrict__ acc, float* __restrict__ out, int n_nodes) {
  const int n = blockIdx.x * blockDim.x + threadIdx.x;
  if (n >= n_nodes) return;
  const float4* p = (const float4*)(acc + (size_t)n * BATCH);
  const float4 a = p[0];
  const float4 b = p[1];
  float s = (a.x + a.y) + (a.z + a.w) + (b.x + b.y) + (b.z + b.w);
  s *= 0.125f;                               // mean over batch (channel dim is 1)
  out[n] = fmaxf(s, 0.0f) + 0.001f;
}

// -------------------------------------------------------------------------
extern "C" void kernel_launch(void* const* d_in, const int* in_sizes, int n_in,
                              void* d_out, int out_size, void* d_ws, size_t ws_size,
                              hipStream_t stream) {
  const float* x   = (const float*)d_in[0];
  const int*   src = (const int*)  d_in[1];
  const int*   dst = (const int*)  d_in[2];
  const float* ew  = (const float*)d_in[3];
  const float* W0  = (const float*)d_in[4];
  const float* b0  = (const float*)d_in[5];
  const float* W1  = (const float*)d_in[6];
  const float* b1  = (const float*)d_in[7];
  const float* W2  = (const float*)d_in[8];
  const float* b2  = (const float*)d_in[9];
  float* out = (float*)d_out;

  const int n_nodes = in_sizes[0] / (BATCH * FDIM);   // 100000
  const int n_edges = in_sizes[1];                    // 3200000
  const int nb      = n_nodes * BATCH;                // 800000

  float* z   = (float*)d_ws;                          // [N, 8]
  float* acc = z + (size_t)nb;                        // [N, 8]  (6.4 MB total)

  const int tiles        = BATCH * (n_nodes / 16);    // 50000 (one per wave)
  const int dense_blocks = (tiles + 7) / 8;           // 8 waves per block
  const int spmm_blocks  = (n_edges + 255) / 256;     // 12500
  const int nb_blocks    = (nb + 255) / 256;          // 3125
  const int fin_blocks   = (n_nodes + 255) / 256;     // 391

  // Layer 1: dense (WMMA, LDS-staged) + init acc, then sparse propagation
  k_dense_wmma<<<dense_blocks, 256, 0, stream>>>(x, W0, b0, z, acc, n_nodes);
  k_spmm<<<spmm_blocks, 256, 0, stream>>>(src, dst, ew, z, acc, n_edges);
  // Layer 2
  k_scale<<<nb_blocks, 256, 0, stream>>>(W1, b1, z, acc, nb);
  k_spmm<<<spmm_blocks, 256, 0, stream>>>(src, dst, ew, z, acc, n_edges);
  // Layer 3
  k_scale<<<nb_blocks, 256, 0, stream>>>(W2, b2, z, acc, nb);
  k_spmm<<<spmm_blocks, 256, 0, stream>>>(src, dst, ew, z, acc, n_edges);
  // Epilogue
  k_final<<<fin_blocks, 256, 0, stream>>>(acc, out, n_nodes);
}